// TeacherGNN_19430432047424
// MI455X (gfx1250) — compile-verified
//
#include <hip/hip_runtime.h>
#include <hip/hip_bf16.h>
#include <math.h>

typedef __attribute__((ext_vector_type(2))) float v2f;
typedef __attribute__((ext_vector_type(8))) float v8f;

#define IN_C  128
#define HID_C 64
#define OUT_C 40

// ---------------- degree / normalization ----------------
__global__ __launch_bounds__(256) void k_set1(float* __restrict__ p, int n) {
  int i = blockIdx.x * blockDim.x + threadIdx.x;
  if (i < n) p[i] = 1.0f;                       // self-loop contributes 1 to deg
}

__global__ __launch_bounds__(256) void k_deg(const int* __restrict__ dst,
                                             float* __restrict__ deg, int E) {
  int e = blockIdx.x * blockDim.x + threadIdx.x;
  if (e < E) atomicAdd(&deg[dst[e]], 1.0f);
}

__global__ __launch_bounds__(256) void k_rsqrt(const float* __restrict__ deg,
                                               float* __restrict__ dis, int n) {
  int i = blockIdx.x * blockDim.x + threadIdx.x;
  if (i < n) dis[i] = rsqrtf(deg[i]);           // deg >= 1 always (self-loop)
}

// ------- per-edge symmetric norm, computed once, reused by all 3 layers -----
__global__ __launch_bounds__(256) void k_edgenorm(const int* __restrict__ src,
                                                  const int* __restrict__ dst,
                                                  const float* __restrict__ dis,
                                                  float* __restrict__ wn, int E) {
  int e = blockIdx.x * blockDim.x + threadIdx.x;
  if (e < E) wn[e] = dis[src[e]] * dis[dst[e]];
}

// ---------------- zero-pad weight matrix columns: W[K,F] -> Wp[K,Fp] --------
__global__ __launch_bounds__(256) void k_padW(const float* __restrict__ W,
                                              float* __restrict__ Wp,
                                              int K, int F, int Fp) {
  int i = blockIdx.x * blockDim.x + threadIdx.x;
  if (i < K * Fp) {
    int k = i / Fp, c = i % Fp;
    Wp[i] = (c < F) ? W[k * F + c] : 0.0f;
  }
}

// ------- fp32 WMMA GEMM: H[N,F] = act(A + bias) @ W[K,Fw(pad)] --------------
// One wave32 computes a 16 x (NT*16) row strip. Per K-step: one b64 A load
// (+ optional fused bias/ReLU on A) reused across NT independent
// V_WMMA_F32_16X16X4_F32 accumulations with distinct B operand registers so
// loads can all issue before the WMMA group drains.
// 32-bit A 16x4 layout: lanes 0-15 M=lane; VGPR0=K+{0|2 by half}, VGPR1=K+{1|3}.
// B 4x16 mirrors it with N=lane&15. C/D: VGPR r -> M = r + 8*half, N=lane&15.
template <int K, int NT, bool FUSEB>
__global__ __launch_bounds__(32) void k_gemm_wmma(const float* __restrict__ A,
                                                  const float* __restrict__ W,
                                                  const float* __restrict__ Bias,
                                                  float* __restrict__ H,
                                                  int N, int Fw, int F) {
  const int lane = threadIdx.x;      // blockDim == 32
  const int hi   = lane >> 4;        // 0: lanes 0-15, 1: lanes 16-31
  const int l15  = lane & 15;
  int row = blockIdx.x * 16 + l15;
  row = (row < N) ? row : (N - 1);   // branchless clamp keeps loads in-bounds
  const float* Ar = A + (size_t)row * K + 2 * hi;       // a = {Ar[k], Ar[k+1]}
  const float* Wc = W + (size_t)(2 * hi) * Fw + l15;    // b = {Wc[k*Fw], Wc[(k+1)*Fw]}

  v8f c[NT];
#pragma unroll
  for (int t = 0; t < NT; ++t) c[t] = (v8f){};

#pragma unroll 4
  for (int k = 0; k < K; k += 4) {
    v2f a = *(const v2f*)(Ar + k);   // one global_load_b64
    if (FUSEB) {                     // fused bias + ReLU of previous layer
      v2f bb = *(const v2f*)(Bias + k + 2 * hi);
      a.x = fmaxf(a.x + bb.x, 0.0f);
      a.y = fmaxf(a.y + bb.y, 0.0f);
    }
    v2f b[NT];
#pragma unroll
    for (int t = 0; t < NT; ++t) {
      b[t].x = Wc[(size_t)(k + 0) * Fw + t * 16];
      b[t].y = Wc[(size_t)(k + 1) * Fw + t * 16];
    }
#pragma unroll
    for (int t = 0; t < NT; ++t)
      c[t] = __builtin_amdgcn_wmma_f32_16x16x4_f32(false, a, false, b[t],
                                                   (short)0, c[t], false, false);
  }

#pragma unroll
  for (int t = 0; t < NT; ++t) {
    int col = t * 16 + l15;
    if (col < F) {
#pragma unroll
      for (int r = 0; r < 8; ++r) {
        int m = blockIdx.x * 16 + r + 8 * hi;
        if (m < N) H[(size_t)m * F + col] = c[t][r];
      }
    }
  }
}

// ---------------- self-loop contribution: AGG = dis[i]^2 * H (float4) -------
// Writes every element -> also serves as accumulator init (ws is poisoned).
__global__ __launch_bounds__(256) void k_selfinit(const float* __restrict__ H,
                                                  const float* __restrict__ dis,
                                                  float* __restrict__ AGG,
                                                  int N, int F) {
  int fchunks = F >> 2;
  long long i = (long long)blockIdx.x * blockDim.x + threadIdx.x;
  if (i < (long long)N * fchunks) {
    float w = dis[(int)(i / fchunks)];
    float ww = w * w;
    float4 h = ((const float4*)H)[i];
    float4 o; o.x = ww * h.x; o.y = ww * h.y; o.z = ww * h.z; o.w = ww * h.w;
    ((float4*)AGG)[i] = o;
  }
}

// ---------------- edge scatter: AGG[dst] += wnorm[e] * H[src] ----------------
// One thread per (edge, 4-feature chunk): float4 gather + 4 global f32 atomics.
// Per-edge norm precomputed once (layer-invariant).
__global__ __launch_bounds__(256) void k_scatter(const int* __restrict__ src,
                                                 const int* __restrict__ dst,
                                                 const float* __restrict__ wn,
                                                 const float* __restrict__ H,
                                                 float* __restrict__ AGG,
                                                 int E, int F) {
  const int chunks = F >> 2;
  long long t = (long long)blockIdx.x * blockDim.x + threadIdx.x;
  if (t >= (long long)E * chunks) return;
  int e  = (int)(t / chunks);
  int cc = (int)(t % chunks);
  int s = src[e], d = dst[e];
  float w = wn[e];
  float4 h = *((const float4*)(H + (size_t)s * F) + cc);
  float* o = AGG + (size_t)d * F + 4 * cc;
  atomicAdd(o + 0, h.x * w);
  atomicAdd(o + 1, h.y * w);
  atomicAdd(o + 2, h.z * w);
  atomicAdd(o + 3, h.w * w);
}

// -------- row-wise log_softmax of (in + bias), F <= 64, one wave32/row ------
__global__ __launch_bounds__(256) void k_log_softmax(const float* __restrict__ in,
                                                     const float* __restrict__ bias,
                                                     float* __restrict__ out,
                                                     int N, int F) {
  int wpb = blockDim.x >> 5;
  int row = blockIdx.x * wpb + (threadIdx.x >> 5);
  if (row >= N) return;
  int lane = threadIdx.x & 31;
  const float* r = in + (size_t)row * F;
  float v0 = (lane      < F) ? r[lane]      + bias[lane]      : -INFINITY;
  float v1 = (lane + 32 < F) ? r[lane + 32] + bias[lane + 32] : -INFINITY;
  float m = fmaxf(v0, v1);
#pragma unroll
  for (int o = 16; o; o >>= 1) m = fmaxf(m, __shfl_xor(m, o, 32));
  float s = ((lane      < F) ? expf(v0 - m) : 0.0f) +
            ((lane + 32 < F) ? expf(v1 - m) : 0.0f);
#pragma unroll
  for (int o = 16; o; o >>= 1) s += __shfl_xor(s, o, 32);
  float lse = m + logf(s);
  if (lane      < F) out[(size_t)row * F + lane]      = v0 - lse;
  if (lane + 32 < F) out[(size_t)row * F + lane + 32] = v1 - lse;
}

static inline int cdiv_ll(long long a, long long b) { return (int)((a + b - 1) / b); }

extern "C" void kernel_launch(void* const* d_in, const int* in_sizes, int n_in,
                              void* d_out, int out_size, void* d_ws, size_t ws_size,
                              hipStream_t stream) {
  const float* x  = (const float*)d_in[0];
  const int*   ei = (const int*)  d_in[1];
  const float* W1 = (const float*)d_in[2];
  const float* b1 = (const float*)d_in[3];
  const float* W2 = (const float*)d_in[4];
  const float* b2 = (const float*)d_in[5];
  const float* W3 = (const float*)d_in[6];
  const float* b3 = (const float*)d_in[7];
  float* out = (float*)d_out;

  const int N = in_sizes[0] / IN_C;
  const int E = in_sizes[1] / 2;
  const int* src = ei;
  const int* dst = ei + E;

  // ws: deg[N] | dis[N] | wnorm[E] | B0[N*64] | B1[N*64] | W3p[64*48] (~59 MB)
  float* deg   = (float*)d_ws;
  float* dis   = deg + N;
  float* wnorm = dis + N;
  float* B0    = wnorm + E;
  float* B1    = B0 + (size_t)N * HID_C;
  float* W3p   = B1 + (size_t)N * HID_C;

  const int OUTP = 48;  // OUT_C padded to multiple of 16

  // ---- normalization (once, reused by all layers) ----
  k_set1    <<<cdiv_ll(N, 256), 256, 0, stream>>>(deg, N);
  k_deg     <<<cdiv_ll(E, 256), 256, 0, stream>>>(dst, deg, E);
  k_rsqrt   <<<cdiv_ll(N, 256), 256, 0, stream>>>(deg, dis, N);
  k_edgenorm<<<cdiv_ll(E, 256), 256, 0, stream>>>(src, dst, dis, wnorm, E);

  const int gN = cdiv_ll(N, 16);

  // ---- layer 1: h1 = x @ W1 ; agg -> B1 (bias/relu deferred to layer 2) ----
  k_gemm_wmma<IN_C, HID_C / 16, false><<<gN, 32, 0, stream>>>(x, W1, nullptr, B0, N, HID_C, HID_C);
  k_selfinit <<<cdiv_ll((long long)N * HID_C / 4, 256), 256, 0, stream>>>(B0, dis, B1, N, HID_C);
  k_scatter  <<<cdiv_ll((long long)E * (HID_C / 4), 256), 256, 0, stream>>>(src, dst, wnorm, B0, B1, E, HID_C);

  // ---- layer 2: h2 = relu(B1 + b1) @ W2 ; agg -> B1 ----
  k_gemm_wmma<HID_C, HID_C / 16, true><<<gN, 32, 0, stream>>>(B1, W2, b1, B0, N, HID_C, HID_C);
  k_selfinit <<<cdiv_ll((long long)N * HID_C / 4, 256), 256, 0, stream>>>(B0, dis, B1, N, HID_C);
  k_scatter  <<<cdiv_ll((long long)E * (HID_C / 4), 256), 256, 0, stream>>>(src, dst, wnorm, B0, B1, E, HID_C);

  // ---- layer 3: h3 = relu(B1 + b2) @ W3p ; agg -> B1 (W3 padded 64x48) ----
  k_padW<<<cdiv_ll(HID_C * OUTP, 256), 256, 0, stream>>>(W3, W3p, HID_C, OUT_C, OUTP);
  k_gemm_wmma<HID_C, OUTP / 16, true><<<gN, 32, 0, stream>>>(B1, W3p, b2, B0, N, OUTP, OUT_C);
  k_selfinit <<<cdiv_ll((long long)N * OUT_C / 4, 256), 256, 0, stream>>>(B0, dis, B1, N, OUT_C);
  k_scatter  <<<cdiv_ll((long long)E * (OUT_C / 4), 256), 256, 0, stream>>>(src, dst, wnorm, B0, B1, E, OUT_C);

  // ---- log_softmax(B1 + b3) -> d_out ----
  k_log_softmax<<<cdiv_ll(N, 256 / 32), 256, 0, stream>>>(B1, b3, out, N, OUT_C);
}